// patch_merging_layer_average_26388279067308
// MI455X (gfx1250) — compile-verified
//
#include <hip/hip_runtime.h>

typedef __attribute__((ext_vector_type(2))) float v2f;
typedef __attribute__((ext_vector_type(8))) float v8f;

#define C_IN   128
#define C_OUT  256
#define N_IN   40962
#define N_OUT  10242
#define BATCH  4
#define NC     (BATCH * N_OUT)   /* 40968 total output columns */
#define NT     32                /* columns per workgroup */
#define SLOTS  32                /* replicated stat buffers */
#define LDS_ST 134               /* padded row stride: conflict-free, 8B aligned */

// ---------------------------------------------------------------------------
// Fused gather-average + GEMM (fp32 WMMA) + partial BN statistics.
// Block: 256 threads (8 waves). Each block handles NT=32 columns, all 256
// output channels. Wave w computes channels [32w, 32w+32).
// ---------------------------------------------------------------------------
__global__ __launch_bounds__(256) void pm_gemm_kernel(
    const float* __restrict__ x, const long long* __restrict__ idx,
    const float* __restrict__ W, float* __restrict__ y,
    float* __restrict__ sums, float* __restrict__ sumsq)
{
  __shared__ float gT[NT][LDS_ST];   // gT[n][c] : transposed g tile
  __shared__ int   poss[NT * 7];

  const int tid  = threadIdx.x;
  const int col0 = blockIdx.x * NT;

  // ---- load the 32x7 gather indices (int64 -> int) ----
  if (tid < NT * 7) {
    int col = col0 + tid / 7;
    int p = 0;
    if (col < NC) {
      int n = col % N_OUT;
      p = (int)idx[(long long)n * 7 + (tid % 7)];
    }
    poss[tid] = p;
  }
  __syncthreads();

  // ---- gather + average 7 neighbors into LDS (transposed) ----
  {
    const int  nn    = tid & (NT - 1);   // column within tile (lane id)
    const int  cg    = tid >> 5;         // 0..7 : channel group
    const int  col   = col0 + nn;
    const bool valid = col < NC;
    const float* xb  = x;
    if (valid) {
      int b = col / N_OUT;
      xb = x + (size_t)b * C_IN * N_IN;
    }
#pragma unroll
    for (int r = 0; r < 16; ++r) {
      int c = cg * 16 + r;
      float s = 0.0f;
      if (valid) {
        const float* xr = xb + (size_t)c * N_IN;
#pragma unroll
        for (int j = 0; j < 7; ++j) s += xr[poss[nn * 7 + j]];
        s *= (1.0f / 7.0f);
      }
      gT[nn][c] = s;   // bank = (6*nn + c) % 64 : conflict-free over 32 lanes
    }
  }
  __syncthreads();

  // ---- fp32 WMMA GEMM: y[32 o x 32 n] per wave, K = 128 ----
  const int lane   = tid & 31;
  const int wv     = tid >> 5;        // wave 0..7
  const int o_base = wv * 32;
  const int m      = lane & 15;       // A: M row / B: N col / C,D: N col
  const int kgrp   = lane >> 4;       // lanes 16-31 hold K=2,3 of each step

  v8f acc00 = {}, acc01 = {}, acc10 = {}, acc11 = {};

  const float* Wr0 = W + (size_t)(o_base + m) * C_IN + kgrp * 2;
  const float* Wr1 = Wr0 + (size_t)16 * C_IN;
  const float* g0  = &gT[m][kgrp * 2];        // n subtile 0 (n = lane&15)
  const float* g1  = &gT[16 + m][kgrp * 2];   // n subtile 1

#pragma unroll 4
  for (int ks = 0; ks < C_IN / 4; ++ks) {
    const int k = ks * 4;
    v2f a0 = *(const v2f*)(Wr0 + k);
    v2f a1 = *(const v2f*)(Wr1 + k);
    v2f b0 = *(const v2f*)(g0 + k);
    v2f b1 = *(const v2f*)(g1 + k);
    acc00 = __builtin_amdgcn_wmma_f32_16x16x4_f32(false, a0, false, b0, (short)0, acc00, false, false);
    acc01 = __builtin_amdgcn_wmma_f32_16x16x4_f32(false, a0, false, b1, (short)0, acc01, false, false);
    acc10 = __builtin_amdgcn_wmma_f32_16x16x4_f32(false, a1, false, b0, (short)0, acc10, false, false);
    acc11 = __builtin_amdgcn_wmma_f32_16x16x4_f32(false, a1, false, b1, (short)0, acc11, false, false);
  }

  // ---- store y (no bias: it cancels in BN) + partial channel stats ----
  const int  ncol0 = col0 + (lane & 15);
  const int  ncol1 = ncol0 + 16;
  const bool val0  = ncol0 < NC;
  const bool val1  = ncol1 < NC;
  int b0i = 0, n0i = 0, b1i = 0, n1i = 0;
  if (val0) { b0i = ncol0 / N_OUT; n0i = ncol0 - b0i * N_OUT; }
  if (val1) { b1i = ncol1 / N_OUT; n1i = ncol1 - b1i * N_OUT; }

  float* sums_s  = sums  + (blockIdx.x % SLOTS) * C_OUT;
  float* sumsq_s = sumsq + (blockIdx.x % SLOTS) * C_OUT;
  const int mgrp = (lane >> 4) * 8;   // C/D: M = vgpr + 8*(lane/16)

#pragma unroll
  for (int so = 0; so < 2; ++so) {
    const v8f A0 = (so == 0) ? acc00 : acc10;
    const v8f A1 = (so == 0) ? acc01 : acc11;
#pragma unroll
    for (int v = 0; v < 8; ++v) {
      const int o  = o_base + so * 16 + mgrp + v;
      const float y0 = A0[v];
      const float y1 = A1[v];
      if (val0) y[((size_t)b0i * C_OUT + o) * N_OUT + n0i] = y0;
      if (val1) y[((size_t)b1i * C_OUT + o) * N_OUT + n1i] = y1;
      float s = (val0 ? y0 : 0.0f) + (val1 ? y1 : 0.0f);
      float q = (val0 ? y0 * y0 : 0.0f) + (val1 ? y1 * y1 : 0.0f);
#pragma unroll
      for (int d = 8; d >= 1; d >>= 1) {   // reduce within 16-lane halves
        s += __shfl_xor(s, d, 32);
        q += __shfl_xor(q, d, 32);
      }
      if ((lane & 15) == 0) {
        atomicAdd(&sums_s[o], s);
        atomicAdd(&sumsq_s[o], q);
      }
    }
  }
}

// ---------------------------------------------------------------------------
__global__ void pm_zero_kernel(float* p, int n) {
  int i = blockIdx.x * blockDim.x + threadIdx.x;
  if (i < n) p[i] = 0.0f;
}

__global__ void pm_finalize_kernel(const float* __restrict__ sums,
                                   const float* __restrict__ sumsq,
                                   const float* __restrict__ gamma,
                                   const float* __restrict__ beta,
                                   float* __restrict__ scale,
                                   float* __restrict__ shift)
{
  const int o = threadIdx.x;   // 256 threads, one per channel
  float s = 0.0f, q = 0.0f;
#pragma unroll
  for (int k = 0; k < SLOTS; ++k) {
    s += sums[k * C_OUT + o];
    q += sumsq[k * C_OUT + o];
  }
  const float inv_cnt = 1.0f / (float)NC;
  const float mu  = s * inv_cnt;
  const float var = q * inv_cnt - mu * mu;
  const float iv  = rsqrtf(var + 1e-5f);
  const float sc  = gamma[o] * iv;
  scale[o] = sc;
  shift[o] = beta[o] - sc * mu;   // conv bias cancels in BN exactly
}

__global__ __launch_bounds__(256) void pm_normalize_kernel(
    float* __restrict__ y, const float* __restrict__ scale,
    const float* __restrict__ shift)
{
  const size_t i = (size_t)blockIdx.x * 256 + threadIdx.x;
  if (i < (size_t)BATCH * C_OUT * N_OUT) {
    const int o = (int)((i / N_OUT) % C_OUT);
    y[i] = scale[o] * y[i] + shift[o];
  }
}

// ---------------------------------------------------------------------------
extern "C" void kernel_launch(void* const* d_in, const int* in_sizes, int n_in,
                              void* d_out, int out_size, void* d_ws, size_t ws_size,
                              hipStream_t stream)
{
  const float*     x     = (const float*)d_in[0];
  const long long* idx   = (const long long*)d_in[1];
  const float*     W     = (const float*)d_in[2];
  /* d_in[3] = conv bias b: cancels in batch-norm, unused */
  const float*     gamma = (const float*)d_in[4];
  const float*     beta  = (const float*)d_in[5];
  float* y  = (float*)d_out;
  float* ws = (float*)d_ws;

  float* sums  = ws;
  float* sumsq = ws + SLOTS * C_OUT;
  float* scale = ws + 2 * SLOTS * C_OUT;
  float* shift = scale + C_OUT;

  const int nstat = 2 * SLOTS * C_OUT;
  pm_zero_kernel<<<(nstat + 255) / 256, 256, 0, stream>>>(ws, nstat);

  const int nblocks = (NC + NT - 1) / NT;   // 1281
  pm_gemm_kernel<<<nblocks, 256, 0, stream>>>(x, idx, W, y, sums, sumsq);

  pm_finalize_kernel<<<1, C_OUT, 0, stream>>>(sums, sumsq, gamma, beta, scale, shift);

  const size_t tot = (size_t)BATCH * C_OUT * N_OUT;
  pm_normalize_kernel<<<(unsigned)((tot + 255) / 256), 256, 0, stream>>>(y, scale, shift);
}